// SignalResize_73358041416198
// MI455X (gfx1250) — compile-verified
//
#include <hip/hip_runtime.h>

#define BB   32
#define TT   8192
#define FF   128
#define OUTT 1024
#define GG   (TT / 16)   // 512 groups of 16 time steps

typedef float v2f __attribute__((ext_vector_type(2)));
typedef float v8f __attribute__((ext_vector_type(8)));

// ---------------------------------------------------------------------------
// Kernel 1: CS16[b,g,f] = sum_{i<16} x[b, 16g+i, f]
// Implemented with V_WMMA_F32_16X16X4_F32, A = all-ones (16x4):
//   D[m,n] = sum_k B[k,n] + C[m,n]  -> every row of D holds the column sums.
// 4 chained WMMAs (K=4 each) sum a 16(time) x 16(feat) tile. Each wave loads
// the 256-float tile exactly once (2 floats per lane per step), so x is
// streamed from HBM exactly once. Column identity only depends on the
// lane<->N striping (lanes {n, n+16} = column n), matching the C/D layout,
// so the result is invariant to the K-slot ordering inside the B layout.
// ---------------------------------------------------------------------------
__global__ __launch_bounds__(256) void k_groupsum_wmma(const float* __restrict__ x,
                                                       float* __restrict__ cs) {
    const int lane = threadIdx.x & 31;
    const int w    = blockIdx.x * 8 + (threadIdx.x >> 5);  // global wave id: 0..131071
    const int fg   = w & 7;               // feature group (8 x 16 = 128)
    const int g    = (w >> 3) & (GG - 1); // time group   (512)
    const int b    = w >> 12;             // batch        (32)

    const int n    = lane & 15;           // column (feature within group)
    const int koff = (lane >> 4) << 1;    // K offset: lanes 0-15 -> {0,1}, 16-31 -> {2,3}

    const float* base = x + ((size_t)b * TT + (size_t)g * 16) * FF + fg * 16 + n;

    v2f a; a.x = 1.0f; a.y = 1.0f;        // all-ones A matrix
    v8f c = {};
#pragma unroll
    for (int k = 0; k < 4; ++k) {
        const float* p = base + (size_t)(4 * k + koff) * FF;
        v2f bb; bb.x = p[0]; bb.y = p[FF];   // two consecutive time steps, same feature
        c = __builtin_amdgcn_wmma_f32_16x16x4_f32(false, a, false, bb, (short)0, c,
                                                  false, false);
    }
    // D row 0: VGPR0, lanes 0-15 hold (M=0, N=lane); all rows equal -> take c[0].
    if (lane < 16)
        cs[((size_t)b * GG + g) * FF + fg * 16 + lane] = c[0];
}

// ---------------------------------------------------------------------------
// Kernel 2: in-place inclusive prefix over the 512 group sums, per (b,f).
// ---------------------------------------------------------------------------
__global__ __launch_bounds__(128) void k_prefix(float* __restrict__ cs) {
    const int f = threadIdx.x;
    const int b = blockIdx.x;
    float run = 0.0f;
    const size_t base = (size_t)b * GG * FF + f;
#pragma unroll 4
    for (int g = 0; g < GG; ++g) {
        const size_t i = base + (size_t)g * FF;
        run += cs[i];
        cs[i] = run;          // inclusive prefix of group sums
    }
}

// ---------------------------------------------------------------------------
// Kernel 3: out[b,t,f] = (ps(t+fsize) - ps(t)) / fsize for t in [0,1024).
// Each thread handles 16 consecutive t for one (b,f). Both endpoint streams
// are contiguous runs of x, seeded from the group-prefix table (hi needs at
// most 15 residual adds), then advanced with one add per output. Re-reads hit
// L2 (x = 128 MB < 192 MB L2).
// ---------------------------------------------------------------------------
__global__ __launch_bounds__(128) void k_out(const float* __restrict__ x,
                                             const float* __restrict__ ps,
                                             const int* __restrict__ lengths,
                                             float* __restrict__ out) {
    const int f    = threadIdx.x;
    const int tile = blockIdx.x;   // 0..63  (16 outputs each)
    const int b    = blockIdx.y;   // 0..31
    const int t0   = tile * 16;

    const int   fsize = lengths[b] - (OUTT - 1);
    const float fsf   = (float)fsize;

    const float* xb = x  + (size_t)b * TT * FF + f;
    const float* pb = ps + (size_t)b * GG * FF + f;

    // lo = exclusive prefix at t0 (t0 is a multiple of 16)
    float lo = (tile == 0) ? 0.0f : pb[(size_t)(tile - 1) * FF];

    // hi = exclusive prefix at idx0 = fsize + t0 (unaligned)
    const int idx0 = fsize + t0;
    const int g0   = idx0 >> 4;
    const int r0   = idx0 & 15;
    float hi = (g0 == 0) ? 0.0f : pb[(size_t)(g0 - 1) * FF];
    const int s0 = idx0 - r0;
    for (int i = 0; i < r0; ++i)
        hi += xb[(size_t)(s0 + i) * FF];

    float* ob = out + ((size_t)b * OUTT + t0) * FF + f;
#pragma unroll
    for (int j = 0; j < 16; ++j) {
        ob[(size_t)j * FF] = (hi - lo) / fsf;
        if (j < 15) {                                  // guard: j=15 advance unused
            lo += xb[(size_t)(t0 + j) * FF];
            hi += xb[(size_t)(idx0 + j) * FF];         // max index fsize+t0+14 <= 8191
        }
    }
}

// ---------------------------------------------------------------------------
extern "C" void kernel_launch(void* const* d_in, const int* in_sizes, int n_in,
                              void* d_out, int out_size, void* d_ws, size_t ws_size,
                              hipStream_t stream) {
    (void)in_sizes; (void)n_in; (void)out_size; (void)ws_size;
    const float* x       = (const float*)d_in[0];
    const int*   lengths = (const int*)d_in[1];
    float*       out     = (float*)d_out;
    float*       cs      = (float*)d_ws;   // 32*512*128 floats = 8 MB scratch

    // 131072 tile-waves / 8 waves per block = 16384 blocks
    k_groupsum_wmma<<<16384, 256, 0, stream>>>(x, cs);
    k_prefix<<<32, 128, 0, stream>>>(cs);
    k_out<<<dim3(64, 32), 128, 0, stream>>>(x, cs, lengths, out);
}